// OosSlidingWindow_5368709120780
// MI455X (gfx1250) — compile-verified
//
#include <hip/hip_runtime.h>

typedef __attribute__((ext_vector_type(2))) float v2f;
typedef __attribute__((ext_vector_type(8))) float v8f;

// Problem constants (from the reference):
//   B=8, S=512, D_C=512, D_G=64, N=8, K_C=64, K_G=8
//   out[b,s, j, r*512 + c*8 + g] = c_s[b,s, j*64+c] * g_s[b,s, r*8+g]
// i.e. per (b,s): outer(c[512], gflat[64]) scattered into 32768 floats.

__global__ __launch_bounds__(256)
void OosSlidingWindow_outer_wmma(const float* __restrict__ c_s,
                                 const float* __restrict__ g_s,
                                 float* __restrict__ out)
{
    const int bs   = blockIdx.x;      // 0..4095  == b*512 + s
    const int tid  = threadIdx.x;     // 0..255   (8 waves, wave32)
    const int wave = tid >> 5;        // 0..7
    const int lane = tid & 31;
    const int half = lane >> 4;       // 0: lanes 0-15, 1: lanes 16-31
    const int ln   = lane & 15;

    const float* __restrict__ crow = c_s + (size_t)bs * 512;
    const float* __restrict__ grow = g_s + (size_t)bs * 64;
    float* __restrict__ obase      = out + (size_t)bs * 32768;

    // ---- B operands: 4 rg-tiles of 16 flattened (r,g) values ----
    // 32-bit B (4x16): row K=0 lives in lanes 0..15 of the first VGPR.
    // Only K=0 is nonzero -> rank-1 product.
    v2f bmat[4];
#pragma unroll
    for (int t = 0; t < 4; ++t) {
        float gv  = grow[t * 16 + ln];          // dup-load, zero upper half
        bmat[t].x = half ? 0.0f : gv;
        bmat[t].y = 0.0f;
    }

    // ---- A operands: this wave's 4 c-tiles of 16 values ----
    // 32-bit A (16x4): column K=0 lives in lanes 0..15 of the first VGPR.
    v2f amat[4];
#pragma unroll
    for (int i = 0; i < 4; ++i) {
        const int ctile = wave * 4 + i;         // 0..31 across the block
        float cv  = crow[ctile * 16 + ln];
        amat[i].x = half ? 0.0f : cv;
        amat[i].y = 0.0f;
    }

    // Precompute this lane's (r,g) pieces for each rg-tile.
#pragma unroll
    for (int i = 0; i < 4; ++i) {
        const int ctile = wave * 4 + i;
        const int j     = ctile >> 2;           // which 64-wide c window
        const int cbase = (ctile & 3) * 16;     // c offset of M=0 within window
        const int m0    = half * 8;             // D layout: lanes 16-31 hold M+8

#pragma unroll
        for (int t = 0; t < 4; ++t) {
            v8f acc = {};
            // D = A x B + 0 : rank-1 outer(c16, g16) in the 16x16 f32 tile.
            v8f d = __builtin_amdgcn_wmma_f32_16x16x4_f32(
                        /*neg_a=*/false, amat[i],
                        /*neg_b=*/false, bmat[t],
                        /*c_mod=*/(short)0, acc,
                        /*reuse_a=*/false, /*reuse_b=*/false);

            // Scatter: element (M=m0+v, N=ln) -> j*4096 + r*512 + c*8 + g
            const int rg = t * 16 + ln;
            const int r  = rg >> 3;
            const int g  = rg & 7;
            float* p = obase + j * 4096 + r * 512 + g;

#pragma unroll
            for (int v = 0; v < 8; ++v) {
                const int cidx = cbase + m0 + v;
                // 512 MiB streamed once, no reuse -> non-temporal (TH=NT)
                __builtin_nontemporal_store(d[v], p + cidx * 8);
            }
        }
    }
}

extern "C" void kernel_launch(void* const* d_in, const int* in_sizes, int n_in,
                              void* d_out, int out_size, void* d_ws, size_t ws_size,
                              hipStream_t stream)
{
    const float* c_s = (const float*)d_in[0];   // [8,512,512] f32
    const float* g_s = (const float*)d_in[1];   // [8,512,64]  f32
    float* out = (float*)d_out;                 // [8,512,8,4096] f32

    (void)in_sizes; (void)n_in; (void)out_size; (void)d_ws; (void)ws_size;

    // One block per (b,s) pair: 4096 blocks x 256 threads (8 wave32 waves).
    OosSlidingWindow_outer_wmma<<<dim3(4096), dim3(256), 0, stream>>>(c_s, g_s, out);
}